// E3PairMix_29609504538684
// MI455X (gfx1250) — compile-verified
//
#include <hip/hip_runtime.h>

// Fused e3nn-style tensor product for MI455X (gfx1250, wave32).
//   w = dist @ W_lin + b_lin        -> V_WMMA_F32_16X16X4_F32 (K=32 = 8 chained), kept in LDS
//   out = sum_p alpha_p w_p CG_p(x,y) -> sparse CG hardcoded, pure VALU FMA
// HBM traffic ~470MB (x+y+out+dist) -> ~20us at 23.3 TB/s; fusing avoids +368MB for w.

typedef float v2f __attribute__((ext_vector_type(2)));
typedef float v8f __attribute__((ext_vector_type(8)));

#define BR     16      // batch rows per block (one WMMA M-tile)
#define NCOLS  352     // C*P = 32*11
#define WSTR   353     // LDS row stride (pad: 8*353 mod 64 = 8 -> conflict-free epilogue)
#define NTILES 22      // 352 / 16
#define KDIM   32      // feature dim F

__global__ __launch_bounds__(256)
void e3_tp_fused(const float* __restrict__ x,
                 const float* __restrict__ y,
                 const float* __restrict__ dist,
                 const float* __restrict__ Wl,
                 const float* __restrict__ bl,
                 float* __restrict__ out,
                 int Btot)
{
    __shared__ float w_lds[BR * WSTR];   // 22.6 KB of the 320 KB WGP LDS

    const int tid    = threadIdx.x;
    const int wv     = tid >> 5;         // wave id (0..7), wave32
    const int lane   = tid & 31;
    const int lh     = lane >> 4;        // lane half
    const int ln     = lane & 15;
    const int b_base = blockIdx.x * BR;

    // ---------------- Phase 1: w = dist @ W_lin + b  (WMMA f32 16x16x4) ------
    // A 16x4 layout: lanes 0-15 -> K=k0+{0,1}, lanes 16-31 -> K=k0+{2,3}; M = ln.
    // B 4x16 layout mirrored: row K = k0 + 2*lh + v, col N = ln.
    int arow_i = b_base + ln; if (arow_i >= Btot) arow_i = Btot - 1;
    const float* arow = dist + (size_t)arow_i * KDIM;

    for (int tn = wv; tn < NTILES; tn += 8) {   // wave-uniform trip count
        const int ncol = tn * 16 + ln;
        v8f acc = {0.f, 0.f, 0.f, 0.f, 0.f, 0.f, 0.f, 0.f};
#pragma unroll
        for (int k0 = 0; k0 < KDIM; k0 += 4) {
            const int ka = k0 + 2 * lh;
            v2f a = *(const v2f*)(arow + ka);            // 8B-aligned (ka even)
            v2f bm;
            bm.x = Wl[(ka    ) * NCOLS + ncol];
            bm.y = Wl[(ka + 1) * NCOLS + ncol];
            acc = __builtin_amdgcn_wmma_f32_16x16x4_f32(
                      false, a, false, bm, (short)0, acc, false, false);
        }
        const float bias = bl[ncol];
#pragma unroll
        for (int r = 0; r < 8; ++r)                      // D row M = r + 8*lh
            w_lds[(r + 8 * lh) * WSTR + ncol] = acc[r] + bias;
    }
    __syncthreads();

    // ---------------- Phase 2: sparse CG tensor product ----------------------
    // irreps: [0:1)=0e scalar, [1:4)=1o vector (m=-1,0,1 ~ y,z,x), [4:9)=2e
    // (m=-2..2 ~ xy,yz,z^2,xz,x^2-y^2).  Constants = alpha(l3) * CG entries:
    const float A0 = 0.57735026919f;   // (0,0,0): sqrt(1/3)
    const float H  = 0.5f;             // (0,1,1)/(1,0,1): sqrt(3/4)/sqrt(3); (0,2,2)/(2,0,2): sqrt(5/4)/sqrt(5)
    const float T  = 0.33333333333f;   // (1,1,0): sqrt(1/3)/sqrt(3)
    const float Q  = 0.25819888975f;   // (2,2,0): sqrt(1/3)/sqrt(5)
    const float Kv = 0.35355339059f;   // (1,1,2): sqrt(5/4)/sqrt(10)
    const float Kw = 0.20412414523f;   // (1,1,2): sqrt(5/4)/sqrt(30)
    const float Jv = 0.27386127875f;   // (1,2,1)/(2,1,1): sqrt(3/4)/sqrt(10)
    const float Jw = 0.15811388301f;   // (1,2,1)/(2,1,1): sqrt(3/4)/sqrt(30)
    const float D1 = 0.26726124191f;   // (2,2,2): sqrt(1/14)
    const float D2 = 0.13363062096f;   // (2,2,2): sqrt(1/14)/2
    const float D3 = 0.23145502495f;   // (2,2,2): sqrt(3/56)

#pragma unroll
    for (int j = 0; j < 2; ++j) {
        const int q  = tid + 256 * j;      // 512 (b,c) pairs per block
        const int bl_ = q >> 5;
        const int c  = q & 31;
        const int b  = b_base + bl_;
        if (b >= Btot) continue;

        const size_t base = ((size_t)b * 32 + c) * 9;
        const float* xp = x + base;
        const float* yp = y + base;

        float X[9], Y[9];
#pragma unroll
        for (int i = 0; i < 9; ++i) { X[i] = xp[i]; Y[i] = yp[i]; }

        const float* wp = &w_lds[bl_ * WSTR + c * 11];
        const float w0 = wp[0], w1 = wp[1],  w2 = wp[2], w3 = wp[3];
        const float w4 = wp[4], w5 = wp[5],  w6 = wp[6], w7 = wp[7];
        const float w8 = wp[8], w9 = wp[9], w10 = wp[10];

        const float x0 = X[0], xv0 = X[1], xv1 = X[2], xv2 = X[3];
        const float xt0 = X[4], xt1 = X[5], xt2 = X[6], xt3 = X[7], xt4 = X[8];
        const float y0 = Y[0], yv0 = Y[1], yv1 = Y[2], yv2 = Y[3];
        const float yt0 = Y[4], yt1 = Y[5], yt2 = Y[6], yt3 = Y[7], yt4 = Y[8];

        // ---- l3 = 0 ----
        const float dvv = xv0*yv0 + xv1*yv1 + xv2*yv2;
        const float dtt = xt0*yt0 + xt1*yt1 + xt2*yt2 + xt3*yt3 + xt4*yt4;
        const float o0 = w0*A0*(x0*y0) + w4*T*dvv + w9*Q*dtt;

        // ---- l3 = 1 : p1,p3 + (1,2,1) S(yt)·xv + (2,1,1) S(xt)·yv ----
        const float Sy00 = -Jw*yt2 - Jv*yt4, Sy11 = 2.f*Jw*yt2, Sy22 = -Jw*yt2 + Jv*yt4;
        const float Sy01 = Jv*yt1, Sy02 = Jv*yt0, Sy12 = Jv*yt3;
        const float Sx00 = -Jw*xt2 - Jv*xt4, Sx11 = 2.f*Jw*xt2, Sx22 = -Jw*xt2 + Jv*xt4;
        const float Sx01 = Jv*xt1, Sx02 = Jv*xt0, Sx12 = Jv*xt3;

        const float o1 = (w1*H)*x0*yv0 + (w3*H)*xv0*y0
                       + w6*(Sy00*xv0 + Sy01*xv1 + Sy02*xv2)
                       + w8*(Sx00*yv0 + Sx01*yv1 + Sx02*yv2);
        const float o2 = (w1*H)*x0*yv1 + (w3*H)*xv1*y0
                       + w6*(Sy01*xv0 + Sy11*xv1 + Sy12*xv2)
                       + w8*(Sx01*yv0 + Sx11*yv1 + Sx12*yv2);
        const float o3 = (w1*H)*x0*yv2 + (w3*H)*xv2*y0
                       + w6*(Sy02*xv0 + Sy12*xv1 + Sy22*xv2)
                       + w8*(Sx02*yv0 + Sx12*yv1 + Sx22*yv2);

        // ---- l3 = 2 : p2,p7 + (1,1,2) + (2,2,2) ----
        const float o4 = (w2*H)*x0*yt0 + (w7*H)*xt0*y0
                       + (w5*Kv)*(xv2*yv0 + xv0*yv2)
                       + w10*(-D1*(xt0*yt2 + xt2*yt0) + D3*(xt1*yt3 + xt3*yt1));
        const float o5 = (w2*H)*x0*yt1 + (w7*H)*xt1*y0
                       + (w5*Kv)*(xv0*yv1 + xv1*yv0)
                       + w10*( D2*(xt1*yt2 + xt2*yt1) - D3*(xt1*yt4 + xt4*yt1)
                             + D3*(xt0*yt3 + xt3*yt0));
        const float o6 = (w2*H)*x0*yt2 + (w7*H)*xt2*y0
                       + (w5*Kw)*(2.f*xv1*yv1 - xv2*yv2 - xv0*yv0)
                       + w10*(-D1*xt0*yt0 + D2*xt1*yt1 + D1*xt2*yt2
                             + D2*xt3*yt3 - D1*xt4*yt4);
        const float o7 = (w2*H)*x0*yt3 + (w7*H)*xt3*y0
                       + (w5*Kv)*(xv2*yv1 + xv1*yv2)
                       + w10*( D2*(xt3*yt2 + xt2*yt3) + D3*(xt3*yt4 + xt4*yt3)
                             + D3*(xt0*yt1 + xt1*yt0));
        const float o8 = (w2*H)*x0*yt4 + (w7*H)*xt4*y0
                       + (w5*Kv)*(xv2*yv2 - xv0*yv0)
                       + w10*(-D1*(xt4*yt2 + xt2*yt4) - D3*xt1*yt1 + D3*xt3*yt3);

        float* op = out + base;
        op[0] = o0; op[1] = o1; op[2] = o2; op[3] = o3; op[4] = o4;
        op[5] = o5; op[6] = o6; op[7] = o7; op[8] = o8;
    }
}

extern "C" void kernel_launch(void* const* d_in, const int* in_sizes, int n_in,
                              void* d_out, int out_size, void* d_ws, size_t ws_size,
                              hipStream_t stream) {
    const float* x  = (const float*)d_in[0];   // [B, 32, 9]
    const float* y  = (const float*)d_in[1];   // [B, 32, 9]
    const float* df = (const float*)d_in[2];   // [B, 32]
    const float* Wl = (const float*)d_in[3];   // [32, 352]
    const float* bl = (const float*)d_in[4];   // [352]
    float* o        = (float*)d_out;           // [B, 32, 9]

    const int Btot = in_sizes[0] / (32 * 9);   // 131072
    const int grid = (Btot + BR - 1) / BR;     // 8192 blocks
    e3_tp_fused<<<grid, 256, 0, stream>>>(x, y, df, Wl, bl, o, Btot);
}